// CausalDenoiser_57526791963183
// MI455X (gfx1250) — compile-verified
//
#include <hip/hip_runtime.h>
#include <hip/hip_bf16.h>

#define NUSERS 100000
#define NITEMS 50000
#define NINTER 1000000
#define NNODES 150000
#define DIM    64
#define NELEM  (NNODES * DIM)          // 9,600,000

typedef __attribute__((ext_vector_type(2))) float v2f;
typedef __attribute__((ext_vector_type(8))) float v8f;

#define ATOMIC_ADD_F32(p, v) \
    __hip_atomic_fetch_add((p), (v), __ATOMIC_RELAXED, __HIP_MEMORY_SCOPE_AGENT)

// ---------------------------------------------------------------- zero fill
__global__ __launch_bounds__(256) void k_zero(float* __restrict__ p, long n) {
    long i = (long)blockIdx.x * blockDim.x + threadIdx.x;
    if (i < n) p[i] = 0.0f;
}

// ------------------------------------------- copy ego -> cur and acc(d_out)
__global__ __launch_bounds__(256) void k_copy(const float4* __restrict__ src,
                                              float4* __restrict__ cur,
                                              float4* __restrict__ acc, long n4) {
    long i = (long)blockIdx.x * blockDim.x + threadIdx.x;
    if (i < n4) { float4 v = src[i]; cur[i] = v; acc[i] = v; }
}

// ---------------------------------------------------- per-node 1/||row||_2
__global__ __launch_bounds__(256) void k_norm(const float* __restrict__ ego,
                                              float* __restrict__ invnorm) {
    int node = blockIdx.x * 8 + (threadIdx.x >> 5);
    if (node >= NNODES) return;
    int lane = threadIdx.x & 31;
    v2f t = *(const v2f*)(ego + (size_t)node * DIM + lane * 2);
    float s = t.x * t.x + t.y * t.y;
    #pragma unroll
    for (int m = 16; m >= 1; m >>= 1) s += __shfl_xor(s, m, 32);
    if (lane == 0) invnorm[node] = 1.0f / fmaxf(sqrtf(s), 1e-12f);
}

// -------- propensity: sim -> e -> loss partial, ipw, degree scatter (8/wave)
#define PSW 8
__global__ __launch_bounds__(256) void k_ps(const float* __restrict__ ego,
                                            const int* __restrict__ uid,
                                            const int* __restrict__ iid,
                                            const float* __restrict__ treat,
                                            const float* __restrict__ alpha_p,
                                            const float* __restrict__ beta_p,
                                            const float* __restrict__ invnorm,
                                            float* __restrict__ ipw,
                                            float* __restrict__ degree,
                                            float* __restrict__ loss) {
    __shared__ float lred[8];
    int wv   = blockIdx.x * 8 + (threadIdx.x >> 5);
    int lane = threadIdx.x & 31;
    float alpha = *alpha_p, beta = *beta_p;
    float lacc = 0.0f;
    long base = (long)wv * PSW;
    for (int it = 0; it < PSW; ++it) {
        long k = base + it;
        if (k >= NINTER) break;
        int u  = uid[k];
        int iN = NUSERS + iid[k];
        v2f du = *(const v2f*)(ego + (size_t)u  * DIM + lane * 2);
        v2f di = *(const v2f*)(ego + (size_t)iN * DIM + lane * 2);
        float p = du.x * di.x + du.y * di.y;
        #pragma unroll
        for (int m = 16; m >= 1; m >>= 1) p += __shfl_xor(p, m, 32);
        if (lane == 0) {
            float sim = p * invnorm[u] * invnorm[iN];
            float z = alpha * sim + beta;
            float e = 1.0f / (1.0f + expf(-z));
            float t = treat[k];
            lacc += t * logf(e) + (1.0f - t) * logf(1.0f - e);
            float w = t / (e + 1e-8f);
            ipw[k] = w;
            ATOMIC_ADD_F32(degree + u,  w);
            ATOMIC_ADD_F32(degree + iN, w);
        }
    }
    if (lane == 0) lred[threadIdx.x >> 5] = lacc;
    __syncthreads();
    if (threadIdx.x == 0) {
        float s = 0.0f;
        #pragma unroll
        for (int i = 0; i < 8; ++i) s += lred[i];
        ATOMIC_ADD_F32(loss, s);
    }
}

// ------------------------------------------- d^-1/2 (in place over degree)
__global__ __launch_bounds__(256) void k_dinv(float* __restrict__ deg, int n) {
    int i = blockIdx.x * blockDim.x + threadIdx.x;
    if (i < n) deg[i] = rsqrtf(deg[i] + 1e-8f);
}

// ------------------------- SpMM scatter: msg += ipw-weighted rows (8/wave)
#define EPW 8
__global__ __launch_bounds__(256) void k_spmm(const float* __restrict__ cur,
                                              const int* __restrict__ uid,
                                              const int* __restrict__ iid,
                                              const float* __restrict__ ipw,
                                              float* __restrict__ msg) {
    long wv   = (long)blockIdx.x * 8 + (threadIdx.x >> 5);
    int  lane = threadIdx.x & 31;
    long base = wv * EPW;
    for (int it = 0; it < EPW; ++it) {
        long k = base + it;
        if (k >= NINTER) return;
        int u  = uid[k];
        int iN = NUSERS + iid[k];
        float w = ipw[k];
        v2f vi = *(const v2f*)(cur + (size_t)iN * DIM + lane * 2);
        v2f vu = *(const v2f*)(cur + (size_t)u  * DIM + lane * 2);
        float* mu = msg + (size_t)u  * DIM + lane * 2;
        float* mi = msg + (size_t)iN * DIM + lane * 2;
        ATOMIC_ADD_F32(mu + 0, w * vi.x);
        ATOMIC_ADD_F32(mu + 1, w * vi.y);
        ATOMIC_ADD_F32(mi + 0, w * vu.x);
        ATOMIC_ADD_F32(mi + 1, w * vu.y);
    }
}

// ---- dense layer via V_WMMA_F32_16X16X4_F32: cur = relu((dinv.*msg)@W^T+b)
// One wave owns 16 rows. A panel (16x64, pre-scaled by d^-1/2) lives in 32
// VGPRs; per N-tile the 16 B v2f's are preloaded into registers so the 16
// loads issue as one clause and the 16 WMMAs run back-to-back. Epilogue is
// fully batched: 32 cur stores as a clause, 32 acc loads as a clause, then
// the adds + acc stores.
__global__ __launch_bounds__(256) void k_dense(const float* __restrict__ msg,
                                               const float* __restrict__ dinv,
                                               const float* __restrict__ W,
                                               const float* __restrict__ bias,
                                               float* __restrict__ cur,
                                               float* __restrict__ acc) {
    int wv = blockIdx.x * 8 + (threadIdx.x >> 5);
    if (wv >= NNODES / 16) return;               // whole-wave exit: EXEC stays all-1s
    int lane = threadIdx.x & 31;
    int m    = lane & 15;                        // row-in-tile / col-in-tile
    int half = lane >> 4;                        // 0: K={0,1}, 1: K={2,3}
    int r0   = wv * 16;
    int row  = r0 + m;
    float dv = dinv[row];

    // A: 16x4 f32 tile layout -> lane<16 holds K=4ks+{0,1}, lane>=16 K=4ks+{2,3}
    v2f A[16];
    const float* mrow = msg + (size_t)row * DIM + 2 * half;
    #pragma unroll
    for (int ks = 0; ks < 16; ++ks) {
        v2f t = *(const v2f*)(mrow + 4 * ks);
        A[ks] = t * dv;
    }

    float outv[4][8];
    #pragma unroll
    for (int nb = 0; nb < 4; ++nb) {
        int n = nb * 16 + m;                     // output column for this lane
        const float* wb = W + (size_t)n * DIM + 2 * half;  // B[k][n] = W[n][k]
        v2f B[16];
        #pragma unroll
        for (int ks = 0; ks < 16; ++ks) B[ks] = *(const v2f*)(wb + 4 * ks);
        v8f c = {};
        #pragma unroll
        for (int ks = 0; ks < 16; ++ks)
            c = __builtin_amdgcn_wmma_f32_16x16x4_f32(
                    false, A[ks], false, B[ks], (short)0, c, false, false);
        float bn = bias[n];
        #pragma unroll
        for (int v = 0; v < 8; ++v)              // D: VGPR v -> row 8*half+v, col m
            outv[nb][v] = fmaxf(c[v] + bn, 0.0f);
    }

    // epilogue (batched clauses, no per-element waits)
    size_t base = (size_t)(r0 + half * 8) * DIM + m;
    #pragma unroll
    for (int v = 0; v < 8; ++v)
        #pragma unroll
        for (int nb = 0; nb < 4; ++nb)
            cur[base + (size_t)v * DIM + nb * 16] = outv[nb][v];

    float a[4][8];
    #pragma unroll
    for (int v = 0; v < 8; ++v)
        #pragma unroll
        for (int nb = 0; nb < 4; ++nb)
            a[nb][v] = acc[base + (size_t)v * DIM + nb * 16];
    #pragma unroll
    for (int v = 0; v < 8; ++v)
        #pragma unroll
        for (int nb = 0; nb < 4; ++nb)
            acc[base + (size_t)v * DIM + nb * 16] = a[nb][v] + outv[nb][v];
}

// --------------------------------------- out = acc/3 ; out[last] = ps_loss
__global__ __launch_bounds__(256) void k_final(float* __restrict__ out,
                                               const float* __restrict__ loss) {
    long i = (long)blockIdx.x * blockDim.x + threadIdx.x;
    if (i < NELEM) out[i] *= (1.0f / 3.0f);
    if (i == 0) out[NELEM] = -loss[0] * (1.0f / (float)NINTER);
}

extern "C" void kernel_launch(void* const* d_in, const int* in_sizes, int n_in,
                              void* d_out, int out_size, void* d_ws, size_t ws_size,
                              hipStream_t stream) {
    const float* ego   = (const float*)d_in[0];
    const int*   uid   = (const int*)  d_in[1];
    const int*   iid   = (const int*)  d_in[2];
    const float* treat = (const float*)d_in[3];
    const float* alpha = (const float*)d_in[4];
    const float* beta  = (const float*)d_in[5];
    const float* W0    = (const float*)d_in[6];
    const float* b0    = (const float*)d_in[7];
    const float* W1    = (const float*)d_in[8];
    const float* b1    = (const float*)d_in[9];
    float* out = (float*)d_out;
    float* ws  = (float*)d_ws;

    // workspace layout (floats)
    const size_t o_loss = 0;
    const size_t o_deg  = 16;
    const size_t o_inv  = o_deg + NNODES;            // 150,016
    const size_t o_ipw  = o_inv + NNODES;            // 300,016
    const size_t o_cur  = o_ipw + NINTER;            // 1,300,016
    const size_t o_msg  = o_cur + (size_t)NELEM;     // 10,900,016  (total ~82 MB)
    float* loss = ws + o_loss;
    float* deg  = ws + o_deg;                        // becomes d^-1/2 after k_dinv
    float* inv  = ws + o_inv;
    float* ipw  = ws + o_ipw;
    float* cur  = ws + o_cur;
    float* msg  = ws + o_msg;

    // 1. zero loss + degree (contiguous prefix)
    k_zero<<<(int)((o_inv + 255) / 256), 256, 0, stream>>>(ws, (long)o_inv);
    // 2. cur = ego ; acc(d_out) = ego
    k_copy<<<(NELEM / 4 + 255) / 256, 256, 0, stream>>>(
        (const float4*)ego, (float4*)cur, (float4*)out, NELEM / 4);
    // 3. per-node inverse norms (1 wave / node)
    k_norm<<<(NNODES + 7) / 8, 256, 0, stream>>>(ego, inv);
    // 4. propensity pass (8 waves/block, 8 interactions/wave)
    k_ps<<<NINTER / (8 * PSW), 256, 0, stream>>>(ego, uid, iid, treat, alpha, beta,
                                                 inv, ipw, deg, loss);
    // 5. d^-1/2
    k_dinv<<<(NNODES + 255) / 256, 256, 0, stream>>>(deg, NNODES);

    // 6. two GCN layers
    const float* Ws[2] = {W0, W1};
    const float* bs[2] = {b0, b1};
    for (int l = 0; l < 2; ++l) {
        k_zero<<<(NELEM + 255) / 256, 256, 0, stream>>>(msg, (long)NELEM);
        k_spmm<<<NINTER / (8 * EPW), 256, 0, stream>>>(cur, uid, iid, ipw, msg);
        k_dense<<<(NNODES / 16 + 7) / 8, 256, 0, stream>>>(msg, deg, Ws[l], bs[l],
                                                           cur, out);
    }
    // 7. denoised = acc/3 ; append ps_loss
    k_final<<<(NELEM + 255) / 256, 256, 0, stream>>>(out, loss);
}